// LightGCN_68564857914047
// MI455X (gfx1250) — compile-verified
//
#include <hip/hip_runtime.h>
#include <hip/hip_bf16.h>
#include <stdint.h>

// ---------------------------------------------------------------------------
// LightGCN propagation for MI455X (gfx1250).
//  - Node table (150002 x 64 f32 = 38.4MB) is L2-resident (192MB L2): gathers
//    and scatter-atomics are L2 traffic, not HBM.
//  - Edge streams (3 x 40MB per layer) are DMA'd into LDS with the Tensor
//    Data Mover (tensor_load_to_lds + s_wait_tensorcnt), double buffered, so
//    the vector-memory pipe is dedicated to the random 256B row gathers and
//    the f32 scatter atomics.
//  - 16 lanes per edge, each lane handles one float4 => each edge's gather is
//    one fully coalesced 256B row read.
//  - acc-update and next-buffer zeroing fused into one sweep per layer.
// ---------------------------------------------------------------------------

#define TILE        2048     // edges per TDM tile (dwords per array)
#define SCAT_BLOCK  256      // 8 waves (wave32)

typedef uint32_t v4u __attribute__((ext_vector_type(4)));
typedef int      v8i __attribute__((ext_vector_type(8)));
typedef int      v4i __attribute__((ext_vector_type(4)));

__device__ __forceinline__ uint32_t lds_byte_offset(const void* p) {
  // Flat LDS addresses carry the LDS byte offset in the low 32 bits
  // (ISA 10.2).
  return (uint32_t)(uintptr_t)p;
}

// Issue one TDM load of `valid_dwords` DWORDs from global `gptr` into LDS at
// byte offset `lds_byte`. Tile is a single line of TILE dwords; reads past
// tensor_dim0 return zero (tail tile). Tracked with TENSORcnt.
__device__ __forceinline__ void tdm_load_dwords(uint32_t lds_byte,
                                                const void* gptr,
                                                unsigned valid_dwords) {
  unsigned long long ga = (unsigned long long)(uintptr_t)gptr;
  unsigned td0 = valid_dwords ? valid_dwords : 1u;

  v4u g0;
  g0[0] = 1u;                                             // count=1, user desc
  g0[1] = lds_byte;                                       // lds_addr
  g0[2] = (uint32_t)(ga & 0xFFFFFFFFull);                 // global_addr[31:0]
  g0[3] = (uint32_t)((ga >> 32) & 0x1FFFFFFull)           // global_addr[56:32]
        | (2u << 30);                                     // type=2 ("image")

  v8i g1;
  g1[0] = (int)(2u << 16);                                // data_size=2 (4B)
  g1[1] = (int)((td0 & 0xFFFFu) << 16);                   // tensor_dim0[15:0]
  g1[2] = (int)(((td0 >> 16) & 0xFFFFu) | (1u << 16));    // td0[31:16], tensor_dim1=1
  g1[3] = (int)(((unsigned)TILE & 0xFFFFu) << 16);        // tile_dim0=TILE
  g1[4] = 1;                                              // tile_dim1=1, tile_dim2=0
  g1[5] = (int)td0;                                       // dim0_stride[31:0]
  g1[6] = (int)((td0 & 0xFFFFu) << 16);                   // dim0_str[47:32]=0, dim1_str[15:0]
  g1[7] = (int)((td0 >> 16) & 0xFFFFu);                   // dim1_stride[47:16]

  v4i z4 = {0, 0, 0, 0};
#if defined(__clang_major__) && (__clang_major__ >= 23)
  v8i z8 = {0, 0, 0, 0, 0, 0, 0, 0};
  __builtin_amdgcn_tensor_load_to_lds(g0, g1, z4, z4, z8, 0);
#else
  __builtin_amdgcn_tensor_load_to_lds(g0, g1, z4, z4, 0);
#endif
}

// ---------------------------------------------------------------------------
// x_a = acc = concat(user_emb, item_emb)   (float4 granularity)
// ---------------------------------------------------------------------------
__global__ void lg_init(const float* __restrict__ ue, const float* __restrict__ ie,
                        float* __restrict__ xa, float* __restrict__ acc,
                        int n_users, int n_nodes) {
  size_t i = (size_t)blockIdx.x * blockDim.x + threadIdx.x;
  size_t n4 = (size_t)n_nodes * 16;       // float4 count (64 floats = 16 f4/row)
  if (i >= n4) return;
  size_t u4 = (size_t)n_users * 16;
  float4 v = (i < u4) ? ((const float4*)ue)[i] : ((const float4*)ie)[i - u4];
  ((float4*)xa)[i]  = v;
  ((float4*)acc)[i] = v;
}

__global__ void lg_zero(float* __restrict__ x, size_t n4) {
  size_t i = (size_t)blockIdx.x * blockDim.x + threadIdx.x;
  if (i >= n4) return;
  ((float4*)x)[i] = make_float4(0.f, 0.f, 0.f, 0.f);
}

// acc += x_new; and zero x_old (the buffer that becomes next layer's output).
__global__ void lg_accadd_zero(float* __restrict__ acc, const float* __restrict__ xn,
                               float* __restrict__ xz, size_t n4) {
  size_t i = (size_t)blockIdx.x * blockDim.x + threadIdx.x;
  if (i >= n4) return;
  float4 a = ((const float4*)acc)[i];
  float4 b = ((const float4*)xn)[i];
  a.x += b.x; a.y += b.y; a.z += b.z; a.w += b.w;
  ((float4*)acc)[i] = a;
  ((float4*)xz)[i]  = make_float4(0.f, 0.f, 0.f, 0.f);
}

// ---------------------------------------------------------------------------
// Scatter layer: x_out[dst] += val * x_in[src] over all edges.
// Edge streams staged into LDS via TDM, double buffered. 16 lanes/edge.
// ---------------------------------------------------------------------------
__global__ void lg_scatter(const int* __restrict__ esrc,
                           const int* __restrict__ edst,
                           const float* __restrict__ evalv,
                           const float* __restrict__ xin,
                           float* __restrict__ xout,
                           int n_edges) {
  __shared__ __align__(16) int   s_src[2][TILE];
  __shared__ __align__(16) int   s_dst[2][TILE];
  __shared__ __align__(16) float s_val[2][TILE];

  const int tid = threadIdx.x;
  const bool issuer = (tid < 32);                 // wave 0 only issues TDM ops
  const int n_tiles = (n_edges + TILE - 1) / TILE;

  uint32_t off_src[2] = { lds_byte_offset(&s_src[0][0]), lds_byte_offset(&s_src[1][0]) };
  uint32_t off_dst[2] = { lds_byte_offset(&s_dst[0][0]), lds_byte_offset(&s_dst[1][0]) };
  uint32_t off_val[2] = { lds_byte_offset(&s_val[0][0]), lds_byte_offset(&s_val[1][0]) };

  const int t0 = blockIdx.x;
  // Prologue: fetch this block's first tile into buffer 0.
  if (issuer && t0 < n_tiles) {
    int start = t0 * TILE;
    unsigned valid = (unsigned)min(TILE, n_edges - start);
    tdm_load_dwords(off_src[0], esrc + start, valid);
    tdm_load_dwords(off_dst[0], edst + start, valid);
    tdm_load_dwords(off_val[0], evalv + start, valid);
  }

  int buf = 0;
  for (int t = t0; t < n_tiles; t += gridDim.x) {
    const int tnext = t + gridDim.x;
    if (issuer) {
      if (tnext < n_tiles) {
        int nstart = tnext * TILE;
        unsigned nvalid = (unsigned)min(TILE, n_edges - nstart);
        int nb = buf ^ 1;
        tdm_load_dwords(off_src[nb], esrc + nstart, nvalid);
        tdm_load_dwords(off_dst[nb], edst + nstart, nvalid);
        tdm_load_dwords(off_val[nb], evalv + nstart, nvalid);
        __builtin_amdgcn_s_wait_tensorcnt(3);   // tile t's 3 DMAs complete
      } else {
        __builtin_amdgcn_s_wait_tensorcnt(0);
      }
    }
    __syncthreads();                            // tile t visible to all waves

    const int start = t * TILE;
    const int valid = min(TILE, n_edges - start);
    const int lane  = tid & 15;                 // float4 slot within 64-f row
    const int grp   = tid >> 4;                 // 16 edge-groups per block

    for (int j = grp; j < valid; j += SCAT_BLOCK / 16) {
      int   s = s_src[buf][j];
      int   d = s_dst[buf][j];
      float v = s_val[buf][j];
      float4 xv = ((const float4*)(xin + (size_t)s * 64))[lane];  // 256B coalesced
      float* o = xout + (size_t)d * 64 + (size_t)lane * 4;
      atomicAdd(o + 0, v * xv.x);
      atomicAdd(o + 1, v * xv.y);
      atomicAdd(o + 2, v * xv.z);
      atomicAdd(o + 3, v * xv.w);
    }
    __syncthreads();                            // done reading buf before reuse
    buf ^= 1;
  }
}

// ---------------------------------------------------------------------------
// out[i] = acc[u[i]] * 0.25   (8192 x 64, float4 granularity)
// ---------------------------------------------------------------------------
__global__ void lg_gather(const float* __restrict__ acc, const int* __restrict__ u,
                          float* __restrict__ out, int n4) {
  int i = blockIdx.x * blockDim.x + threadIdx.x;
  if (i >= n4) return;
  int row = i >> 4;
  int q   = i & 15;
  int node = u[row];
  float4 v = ((const float4*)(acc + (size_t)node * 64))[q];
  v.x *= 0.25f; v.y *= 0.25f; v.z *= 0.25f; v.w *= 0.25f;
  ((float4*)out)[i] = v;
}

// ---------------------------------------------------------------------------
extern "C" void kernel_launch(void* const* d_in, const int* in_sizes, int n_in,
                              void* d_out, int out_size, void* d_ws, size_t ws_size,
                              hipStream_t stream) {
  const float* user_emb = (const float*)d_in[0];
  const float* item_emb = (const float*)d_in[1];
  const int*   esrc     = (const int*)d_in[2];
  const int*   edst     = (const int*)d_in[3];
  const float* evalv    = (const float*)d_in[4];
  const int*   uidx     = (const int*)d_in[5];

  const int D       = 64;
  const int n_users = in_sizes[0] / D;
  const int n_items = in_sizes[1] / D;
  const int n_edges = in_sizes[2];
  const int n_u     = in_sizes[5];
  const int n_nodes = n_users + n_items;

  // Workspace: [x_a | x_b | acc], each n_nodes*64 f32 (38.4MB) -> 115MB total.
  size_t per = (size_t)n_nodes * D;
  float* xa  = (float*)d_ws;
  float* xb  = xa + per;
  float* acc = xb + per;

  const size_t n4    = per / 4;                       // float4 elements
  const int    eb    = 256;
  const int    egrid = (int)((n4 + eb - 1) / eb);

  lg_init<<<egrid, eb, 0, stream>>>(user_emb, item_emb, xa, acc, n_users, n_nodes);
  lg_zero<<<egrid, eb, 0, stream>>>(xb, n4);          // only zero pass needed

  const int n_tiles = (n_edges + TILE - 1) / TILE;
  const int sgrid   = n_tiles < 1024 ? n_tiles : 1024;

  for (int layer = 0; layer < 3; ++layer) {
    lg_scatter<<<sgrid, SCAT_BLOCK, 0, stream>>>(esrc, edst, evalv, xa, xb, n_edges);
    // acc += xb; zero xa so it can serve as next layer's scatter output.
    lg_accadd_zero<<<egrid, eb, 0, stream>>>(acc, xb, xa, n4);
    float* t = xa; xa = xb; xb = t;                   // next layer input = new x
  }

  const int g4 = n_u * 16;                            // 8192*64/4 float4s
  lg_gather<<<(g4 + 255) / 256, 256, 0, stream>>>(acc, uidx, (float*)d_out, g4);
}